// LiftFeatLoss_58274116272578
// MI455X (gfx1250) — compile-verified
//
#include <hip/hip_runtime.h>
#include <cfloat>
#include <math.h>

// Problem constants (from reference setup_inputs)
#define B_   8
#define C_   128
#define H8_  120
#define W8_  160
#define H_   960
#define W_   1280
#define N_   2048
#define TEMP_ 0.2f

#define SEG_       4                    // column-tile segments per row tile
#define TILES_     (N_ / 16)            // 128 column tiles
#define TPS_       (TILES_ / SEG_)      // 32 tiles per segment

typedef float v8f __attribute__((ext_vector_type(8)));
typedef float v2f __attribute__((ext_vector_type(2)));

// ---------------------------------------------------------------------------
// Curvature helpers (depthwise Sobel, zero padding, 3 channels)
// ---------------------------------------------------------------------------
__device__ __forceinline__ float ld_pix(const float* __restrict__ pc, int y, int x) {
    return (y >= 0 && y < H_ && x >= 0 && x < W_) ? pc[(size_t)y * W_ + x] : 0.0f;
}

__device__ __forceinline__ float curv_at(const float* __restrict__ p, int y, int x) {
    float acc = 0.0f;
#pragma unroll
    for (int c = 0; c < 3; ++c) {
        const float* pc = p + (size_t)c * H_ * W_;
        float n00 = ld_pix(pc, y - 1, x - 1);
        float n01 = ld_pix(pc, y - 1, x);
        float n02 = ld_pix(pc, y - 1, x + 1);
        float n10 = ld_pix(pc, y,     x - 1);
        float n12 = ld_pix(pc, y,     x + 1);
        float n20 = ld_pix(pc, y + 1, x - 1);
        float n21 = ld_pix(pc, y + 1, x);
        float n22 = ld_pix(pc, y + 1, x + 1);
        float gx = (n02 - n00) + 2.0f * (n12 - n10) + (n22 - n20);
        float gy = (n20 - n00) + 2.0f * (n21 - n01) + (n22 - n02);
        acc += gx * gx + gy * gy;
    }
    return sqrtf(acc);
}

// ---------------------------------------------------------------------------
// Kernel 0: init per-image min/max (uint-encoded nonnegative floats)
// ---------------------------------------------------------------------------
__global__ void init_minmax_kernel(unsigned int* __restrict__ mm) {
    int t = threadIdx.x;
    if (t < 2 * B_ * 2) mm[t] = (t & 1) ? 0u : 0x7F7FFFFFu;  // max=0, min=FLT_MAX
}

// ---------------------------------------------------------------------------
// Kernel 1: per-image curvature min/max reduction
// ---------------------------------------------------------------------------
__global__ void curv_minmax_kernel(const float* __restrict__ normals,
                                   unsigned int* __restrict__ mm) {
    int b = blockIdx.y;
    const float* p = normals + (size_t)b * 3 * H_ * W_;
    float lmin = FLT_MAX, lmax = 0.0f;
    int total = H_ * W_;
    for (int idx = blockIdx.x * blockDim.x + threadIdx.x; idx < total;
         idx += gridDim.x * blockDim.x) {
        int y = idx / W_, x = idx - y * W_;
        float c = curv_at(p, y, x);
        lmin = fminf(lmin, c);
        lmax = fmaxf(lmax, c);
    }
    __shared__ float smin[256], smax[256];
    int t = threadIdx.x;
    smin[t] = lmin; smax[t] = lmax;
    __syncthreads();
    for (int off = 128; off > 0; off >>= 1) {
        if (t < off) {
            smin[t] = fminf(smin[t], smin[t + off]);
            smax[t] = fmaxf(smax[t], smax[t + off]);
        }
        __syncthreads();
    }
    if (t == 0) {
        atomicMin(&mm[b * 2 + 0], __float_as_uint(smin[0]));
        atomicMax(&mm[b * 2 + 1], __float_as_uint(smax[0]));
    }
}

// ---------------------------------------------------------------------------
// Kernel 2: normalize curvature -> w_kpt, 8x8 mean pool -> (B,120,160)
// ---------------------------------------------------------------------------
__global__ void pool_w8_kernel(const float* __restrict__ normals,
                               const unsigned int* __restrict__ mm,
                               float* __restrict__ ws8) {
    int x8 = blockIdx.x, y8 = blockIdx.y, b = blockIdx.z;
    int t = threadIdx.x;
    int py = t >> 3, px = t & 7;
    const float* p = normals + (size_t)b * 3 * H_ * W_;
    float mn = __uint_as_float(mm[b * 2 + 0]);
    float mx = __uint_as_float(mm[b * 2 + 1]);
    float rng = mx - mn;
    float c = curv_at(p, y8 * 8 + py, x8 * 8 + px);
    float cn = (rng > 1e-7f) ? (c - mn) / (rng + 1e-7f) : 0.0f;
    float w = 1.0f + 2.0f * cn;
    __shared__ float sh[64];
    sh[t] = w;
    __syncthreads();
    for (int off = 32; off > 0; off >>= 1) {
        if (t < off) sh[t] += sh[t + off];
        __syncthreads();
    }
    if (t == 0) ws8[((size_t)b * H8_ + y8) * W8_ + x8] = sh[0] * (1.0f / 64.0f);
}

// ---------------------------------------------------------------------------
// Kernel 3: gather descriptors at keypoints + pair weights
// ---------------------------------------------------------------------------
__global__ void gather_kernel(const float* __restrict__ d1, const float* __restrict__ d2,
                              const int* __restrict__ pts,
                              const float* __restrict__ w1s8, const float* __restrict__ w2s8,
                              float* __restrict__ m1, float* __restrict__ m2,
                              float* __restrict__ pw) {
    int n = blockIdx.x, b = blockIdx.y, c = threadIdx.x;
    const int* q = pts + ((size_t)b * N_ + n) * 4;
    int x1 = q[0], y1 = q[1], x2 = q[2], y2 = q[3];
    size_t o = ((size_t)b * N_ + n) * C_;
    m1[o + c] = d1[(((size_t)b * C_ + c) * H8_ + y1) * W8_ + x1];
    m2[o + c] = d2[(((size_t)b * C_ + c) * H8_ + y2) * W8_ + x2];
    if (c == 0) {
        pw[(size_t)b * N_ + n] =
            0.5f * (w1s8[((size_t)b * H8_ + y1) * W8_ + x1] +
                    w2s8[((size_t)b * H8_ + y2) * W8_ + x2]);
    }
}

// ---------------------------------------------------------------------------
// Kernel 4: WMMA fp32 GEMM segment pass with per-lane online softmax.
// One wave per (16-row tile, column segment). All 32 B fragments of a tile
// are preloaded into distinct VGPRs -> 32 batched global_load_b64, one wait,
// 32 dense v_wmma_f32_16x16x4_f32. Per-lane (m,s) online softmax keeps the
// hot loop shuffle-free; shuffle reductions happen once per segment.
// ---------------------------------------------------------------------------
__global__ __launch_bounds__(32) void dist_pass_kernel(
    const float* __restrict__ Am, const float* __restrict__ Bm,
    float* __restrict__ seg_m, float* __restrict__ seg_s,
    float* __restrict__ seg_bv, int* __restrict__ seg_bi,
    float* __restrict__ diag, int store_aux) {
    const int rt   = blockIdx.x;           // row tile 0..TILES_-1
    const int b    = blockIdx.y;
    const int sg   = blockIdx.z;           // column segment 0..SEG_-1
    const int lane = threadIdx.x;          // 0..31 (wave32)
    const int m    = lane & 15;
    const int h    = lane >> 4;
    const int kk   = h * 2;                // K sub-offset for f32 16x16x4 fragments
    const int rowbase = rt * 16;

    const float* A  = Am + (size_t)b * N_ * C_;
    const float* Bp = Bm + (size_t)b * N_ * C_;

    // Preload all 32 A k-fragments for this row tile (K = 128 = 32 * 4).
    v2f aF[32];
    {
        const float* arow = A + (size_t)(rowbase + m) * C_ + kk;
#pragma unroll
        for (int kt = 0; kt < 32; ++kt)
            aF[kt] = *(const v2f*)(arow + kt * 4);
    }

    // Per-lane online softmax state, 8 rows per lane (rows r+8h, r=0..7).
    float run_m[8], run_s[8], bestv[8];
    int besti[8];
#pragma unroll
    for (int r = 0; r < 8; ++r) {
        run_m[r] = -INFINITY; run_s[r] = 0.0f;
        bestv[r] = -INFINITY; besti[r] = 0;
    }

    const int ct0 = sg * TPS_;
    for (int ci = 0; ci < TPS_; ++ci) {
        const int ct = ct0 + ci;
        const int colbase = ct * 16;
        const float* brow = Bp + (size_t)(colbase + m) * C_ + kk;
        if (ci + 1 < TPS_)
            __builtin_prefetch(Bp + (size_t)(colbase + 16 + m) * C_, 0, 1);

        // Batched fragment loads: 32 distinct destinations -> one wait total.
        v2f bF[32];
#pragma unroll
        for (int kt = 0; kt < 32; ++kt)
            bF[kt] = *(const v2f*)(brow + kt * 4);

        v8f acc = {0.0f, 0.0f, 0.0f, 0.0f, 0.0f, 0.0f, 0.0f, 0.0f};
#pragma unroll
        for (int kt = 0; kt < 32; ++kt)
            acc = __builtin_amdgcn_wmma_f32_16x16x4_f32(
                false, aF[kt], false, bF[kt], (short)0, acc, false, false);

        const int col = colbase + m;
#pragma unroll
        for (int r = 0; r < 8; ++r) {
            float v = acc[r] * TEMP_;
            if (v > bestv[r]) { bestv[r] = v; besti[r] = col; }
            float nm = fmaxf(run_m[r], v);
            run_s[r] = run_s[r] * __expf(run_m[r] - nm) + __expf(v - nm);
            run_m[r] = nm;
            if (store_aux && ct == rt) {
                if (m == r + 8 * h)
                    diag[(size_t)b * N_ + rowbase + r + 8 * h] = v;
            }
        }
    }

    // Segment finalize: merge per-lane states across each 16-lane half.
#pragma unroll
    for (int r = 0; r < 8; ++r) {
        float M = run_m[r];
        M = fmaxf(M, __shfl_xor(M, 1));
        M = fmaxf(M, __shfl_xor(M, 2));
        M = fmaxf(M, __shfl_xor(M, 4));
        M = fmaxf(M, __shfl_xor(M, 8));
        float s = run_s[r] * __expf(run_m[r] - M);
        s += __shfl_xor(s, 1);
        s += __shfl_xor(s, 2);
        s += __shfl_xor(s, 4);
        s += __shfl_xor(s, 8);
        float bv = bestv[r];
        int   bi = besti[r];
#pragma unroll
        for (int msk = 1; msk <= 8; msk <<= 1) {
            float ov = __shfl_xor(bv, msk);
            int   oi = __shfl_xor(bi, msk);
            if (ov > bv || (ov == bv && oi < bi)) { bv = ov; bi = oi; }
        }
        int row = rowbase + r + 8 * h;
        size_t o = ((size_t)sg * B_ + b) * N_ + row;
        if (m == 0) {
            seg_m[o]  = M;
            seg_s[o]  = s;
            seg_bv[o] = bv;
            seg_bi[o] = bi;
        }
    }
}

// ---------------------------------------------------------------------------
// Kernel 5: merge per-segment partial softmax stats -> per-row (max, sum),
// argmax-correct flag.
// ---------------------------------------------------------------------------
__global__ void merge_kernel(const float* __restrict__ seg_m, const float* __restrict__ seg_s,
                             const float* __restrict__ seg_bv, const int* __restrict__ seg_bi,
                             float* __restrict__ out_m, float* __restrict__ out_s,
                             float* __restrict__ correct, int store_aux) {
    int i = blockIdx.x * blockDim.x + threadIdx.x;   // 0..B_*N_-1
    if (i >= B_ * N_) return;
    int b = i / N_;
    int row = i - b * N_;
    float M = -INFINITY;
#pragma unroll
    for (int s = 0; s < SEG_; ++s)
        M = fmaxf(M, seg_m[((size_t)s * B_ + b) * N_ + row]);
    float S = 0.0f;
    float bv = -INFINITY;
    int bi = 0x7FFFFFFF;
#pragma unroll
    for (int s = 0; s < SEG_; ++s) {
        size_t o = ((size_t)s * B_ + b) * N_ + row;
        S += seg_s[o] * __expf(seg_m[o] - M);
        float ov = seg_bv[o];
        int   oi = seg_bi[o];
        if (ov > bv || (ov == bv && oi < bi)) { bv = ov; bi = oi; }
    }
    out_m[i] = M;
    out_s[i] = S;
    if (store_aux) correct[i] = (bi == row) ? 1.0f : 0.0f;
}

// ---------------------------------------------------------------------------
// Kernel 6: per-image combine -> loss_b, acc_b, conf
// ---------------------------------------------------------------------------
__global__ void combine_kernel(const float* __restrict__ row_m, const float* __restrict__ row_s,
                               const float* __restrict__ col_m, const float* __restrict__ col_s,
                               const float* __restrict__ diag, const float* __restrict__ correct,
                               const float* __restrict__ pw,
                               float* __restrict__ conf_out, float* __restrict__ la) {
    int b = blockIdx.x;
    int t = threadIdx.x;
    float sw = 0.0f, swl = 0.0f, sl = 0.0f, sa = 0.0f;
    for (int i = t; i < N_; i += 256) {
        size_t o = (size_t)b * N_ + i;
        float d    = diag[o];
        float rs   = row_s[o];
        float cs   = col_s[o];
        float lp12 = d - row_m[o] - __logf(rs);
        float lp21 = d - col_m[o] - __logf(cs);
        float lpp  = -0.5f * (lp12 + lp21);
        float w    = pw[o];
        sw  += w;
        swl += lpp * w;
        sl  += lpp;
        sa  += correct[o];
        conf_out[o] = 1.0f / (rs * cs);
    }
    __shared__ float sh[4][256];
    sh[0][t] = sw; sh[1][t] = swl; sh[2][t] = sl; sh[3][t] = sa;
    __syncthreads();
    for (int off = 128; off > 0; off >>= 1) {
        if (t < off) {
            sh[0][t] += sh[0][t + off];
            sh[1][t] += sh[1][t + off];
            sh[2][t] += sh[2][t + off];
            sh[3][t] += sh[3][t + off];
        }
        __syncthreads();
    }
    if (t == 0) {
        float SW = sh[0][0], SWL = sh[1][0], SL = sh[2][0], SA = sh[3][0];
        la[b]      = (SW > 1e-8f) ? (SWL / SW) : (SL / (float)N_);
        la[B_ + b] = SA / (float)N_;
    }
}

// ---------------------------------------------------------------------------
// Kernel 7: final mean over images -> out[0]=loss, out[1]=acc
// ---------------------------------------------------------------------------
__global__ void final_kernel(const float* __restrict__ la, float* __restrict__ out) {
    if (threadIdx.x == 0) {
        float l = 0.0f, a = 0.0f;
        for (int b = 0; b < B_; ++b) { l += la[b]; a += la[B_ + b]; }
        out[0] = l * (1.0f / B_);
        out[1] = a * (1.0f / B_);
    }
}

// ---------------------------------------------------------------------------
extern "C" void kernel_launch(void* const* d_in, const int* in_sizes, int n_in,
                              void* d_out, int out_size, void* d_ws, size_t ws_size,
                              hipStream_t stream) {
    const float* descs1   = (const float*)d_in[0];
    const float* descs2   = (const float*)d_in[1];
    const float* normals1 = (const float*)d_in[2];
    const float* normals2 = (const float*)d_in[3];
    const int*   pts      = (const int*)d_in[4];
    float* out = (float*)d_out;  // [loss, acc, conf(8*2048)]

    // Workspace layout (floats)
    float* ws = (float*)d_ws;
    const size_t BNC = (size_t)B_ * N_ * C_;      // 2,097,152
    const size_t BN  = (size_t)B_ * N_;           // 16,384
    const size_t BS8 = (size_t)B_ * H8_ * W8_;    // 153,600
    float* m1    = ws;
    float* m2    = m1 + BNC;
    float* pw    = m2 + BNC;
    float* row_m = pw + BN;
    float* row_s = row_m + BN;
    float* col_m = row_s + BN;
    float* col_s = col_m + BN;
    float* diag  = col_s + BN;
    float* corr  = diag + BN;
    float* w1s8  = corr + BN;
    float* w2s8  = w1s8 + BS8;
    float* la    = w2s8 + BS8;                    // 2*B_ per-image loss/acc
    float* seg_m  = la + 2 * B_;                  // SEG_*BN each
    float* seg_s  = seg_m + SEG_ * BN;
    float* seg_bv = seg_s + SEG_ * BN;
    int*   seg_bi = (int*)(seg_bv + SEG_ * BN);
    unsigned int* mm = (unsigned int*)(seg_bi + SEG_ * BN);  // 2 imgs * B * {min,max}

    // Phase 1: curvature weights (memory-bound stencil; L2-resident)
    init_minmax_kernel<<<1, 64, 0, stream>>>(mm);
    dim3 gmm(512, B_);
    curv_minmax_kernel<<<gmm, 256, 0, stream>>>(normals1, mm);
    curv_minmax_kernel<<<gmm, 256, 0, stream>>>(normals2, mm + 2 * B_);
    dim3 gpool(W8_, H8_, B_);
    pool_w8_kernel<<<gpool, 64, 0, stream>>>(normals1, mm, w1s8);
    pool_w8_kernel<<<gpool, 64, 0, stream>>>(normals2, mm + 2 * B_, w2s8);

    // Phase 2: keypoint gather
    dim3 ggat(N_, B_);
    gather_kernel<<<ggat, C_, 0, stream>>>(descs1, descs2, pts, w1s8, w2s8, m1, m2, pw);

    // Phase 3: WMMA GEMM + streaming dual softmax (segmented, then merged)
    dim3 gpass(TILES_, B_, SEG_);
    dim3 gmerge((unsigned)((BN + 255) / 256));
    dist_pass_kernel<<<gpass, 32, 0, stream>>>(m1, m2, seg_m, seg_s, seg_bv, seg_bi,
                                               diag, 1);
    merge_kernel<<<gmerge, 256, 0, stream>>>(seg_m, seg_s, seg_bv, seg_bi,
                                             row_m, row_s, corr, 1);
    dist_pass_kernel<<<gpass, 32, 0, stream>>>(m2, m1, seg_m, seg_s, seg_bv, seg_bi,
                                               diag, 0);
    merge_kernel<<<gmerge, 256, 0, stream>>>(seg_m, seg_s, seg_bv, seg_bi,
                                             col_m, col_s, corr, 0);

    // Phase 4: combine + final reduction
    combine_kernel<<<B_, 256, 0, stream>>>(row_m, row_s, col_m, col_s, diag, corr, pw,
                                           out + 2, la);
    final_kernel<<<1, 32, 0, stream>>>(la, out);
}